// GCN_86887188398501
// MI455X (gfx1250) — compile-verified
//
#include <hip/hip_runtime.h>
#include <math.h>

#define D 128

typedef __attribute__((ext_vector_type(2))) float v2f;
typedef __attribute__((ext_vector_type(8))) float v8f;

// ---------------- utility ----------------
__global__ void gcn_zero_f32(float* p, long n) {
    long i = (long)blockIdx.x * blockDim.x + threadIdx.x;
    long stride = (long)gridDim.x * blockDim.x;
    for (; i < n; i += stride) p[i] = 0.0f;
}

// deg[dst[e]] += 1  (self-loop added in make_dinv)
__global__ void gcn_degree(const int* __restrict__ dst, float* __restrict__ deg, int nE) {
    int e = blockIdx.x * blockDim.x + threadIdx.x;
    if (e < nE) atomicAdd(&deg[dst[e]], 1.0f);
}

__global__ void gcn_make_dinv(float* deg, int n) {
    int i = blockIdx.x * blockDim.x + threadIdx.x;
    if (i < n) deg[i] = rsqrtf(deg[i] + 1.0f);   // +1 self loop; always > 0
}

// ---------------- fp32 WMMA GEMM: C[N x 128] = A[N x 128] * W[128 x 128] ----------------
// One block = 256 threads = 8 waves. Block handles one 16-row tile; wave w does 16-col tile w.
__global__ __launch_bounds__(256) void gcn_gemm128(const float* __restrict__ A,
                                                   const float* __restrict__ W,
                                                   float* __restrict__ C) {
    const int wave = threadIdx.x >> 5;
    const int lane = threadIdx.x & 31;
    const int row_base = blockIdx.x * 16;
    const int col_base = wave * 16;

    const int r     = lane & 15;       // row-in-tile for A, col-in-tile for B
    const int khalf = lane >> 4;       // selects K pair {0,1} or {2,3}

    v8f acc = {};
    #pragma unroll 4
    for (int k0 = 0; k0 < D; k0 += 4) {
        const int ka = k0 + khalf * 2;
        // A fragment: lane holds A[row][ka], A[row][ka+1]
        const float2 a2 = *(const float2*)(A + (size_t)(row_base + r) * D + ka);
        v2f av; av.x = a2.x; av.y = a2.y;
        // B fragment: lane holds W[ka][col], W[ka+1][col]
        const float b0 = W[(size_t)(ka + 0) * D + col_base + r];
        const float b1 = W[(size_t)(ka + 1) * D + col_base + r];
        v2f bv; bv.x = b0; bv.y = b1;
        acc = __builtin_amdgcn_wmma_f32_16x16x4_f32(false, av, false, bv,
                                                    (short)0, acc, false, false);
    }
    // D layout: lanes 0-15 -> N=lane, M=vgpr; lanes 16-31 -> N=lane-16, M=vgpr+8
    const int n = col_base + (lane & 15);
    const int mbase = row_base + (lane >> 4) * 8;
    #pragma unroll
    for (int i = 0; i < 8; ++i)
        C[(size_t)(mbase + i) * D + n] = acc[i];
}

// ---------------- edge scatter: agg[dst] += H[src] * dinv[src]*dinv[dst] ----------------
// One wave per edge; each lane handles 4 columns (float4).
__global__ __launch_bounds__(256) void gcn_scatter(const float* __restrict__ H,
                                                   const int* __restrict__ src,
                                                   const int* __restrict__ dst,
                                                   const float* __restrict__ dinv,
                                                   float* __restrict__ agg, int nE) {
    const int e = blockIdx.x * (blockDim.x >> 5) + (threadIdx.x >> 5);
    if (e >= nE) return;
    const int lane = threadIdx.x & 31;
    const int s = src[e], d = dst[e];
    const float nrm = dinv[s] * dinv[d];
    const float4 v = ((const float4*)(H + (size_t)s * D))[lane];
    float* o = agg + (size_t)d * D + lane * 4;
    atomicAdd(o + 0, v.x * nrm);
    atomicAdd(o + 1, v.y * nrm);
    atomicAdd(o + 2, v.z * nrm);
    atomicAdd(o + 3, v.w * nrm);
}

// out = agg + H*dinv^2 + bias   (self-loop term + bias), in-place safe
__global__ void gcn_combine(const float* __restrict__ agg, const float* __restrict__ H,
                            const float* __restrict__ dinv, const float* __restrict__ bias,
                            float* __restrict__ out, long nElem) {
    long i = (long)blockIdx.x * blockDim.x + threadIdx.x;
    long stride = (long)gridDim.x * blockDim.x;
    for (; i < nElem; i += stride) {
        const int row = (int)(i >> 7);
        const int col = (int)(i & (D - 1));
        const float di = dinv[row];
        out[i] = agg[i] + H[i] * di * di + bias[col];
    }
}

// ---------------- batch norm ----------------
#define BN_ROWS 256
__global__ __launch_bounds__(128) void gcn_bn_stats(const float* __restrict__ x,
                                                    float* __restrict__ stats, int nNodes) {
    const int col = threadIdx.x;
    const int r0 = blockIdx.x * BN_ROWS;
    const int rend = min(r0 + BN_ROWS, nNodes);
    float s = 0.0f, sq = 0.0f;
    for (int r = r0; r < rend; ++r) {
        const float v = x[(size_t)r * D + col];
        s += v; sq += v * v;
    }
    atomicAdd(&stats[col], s);
    atomicAdd(&stats[D + col], sq);
}

__global__ __launch_bounds__(128) void gcn_bn_final(const float* __restrict__ stats,
                                                    const float* __restrict__ gamma,
                                                    const float* __restrict__ beta,
                                                    const float* __restrict__ bs,
                                                    const float* __restrict__ bd,
                                                    float* __restrict__ bnscale,
                                                    float* __restrict__ bnshift,
                                                    float* __restrict__ bsum, int nNodes) {
    const int c = threadIdx.x;
    const float invn = 1.0f / (float)nNodes;
    const float mean = stats[c] * invn;
    const float var  = stats[D + c] * invn - mean * mean;
    const float sc   = gamma[c] * rsqrtf(var + 1e-5f);
    bnscale[c] = sc;
    bnshift[c] = beta[c] - mean * sc;
    bsum[c] = bs[c] + bd[c];
}

__global__ void gcn_bn_apply(float* __restrict__ x, const float* __restrict__ bnscale,
                             const float* __restrict__ bnshift, long nElem) {
    long i = (long)blockIdx.x * blockDim.x + threadIdx.x;
    long stride = (long)gridDim.x * blockDim.x;
    for (; i < nElem; i += stride) {
        const int col = (int)(i & (D - 1));
        x[i] = fmaxf(x[i] * bnscale[col] + bnshift[col], 0.0f);
    }
}

// ---------------- decoder: out[e] = sigmoid(relu(P[a]+Q[b]+bsum) . Wf + bf) ----------------
__global__ __launch_bounds__(256) void gcn_decode(const float* __restrict__ P,
                                                  const float* __restrict__ Q,
                                                  const int* __restrict__ ia,
                                                  const int* __restrict__ ib,
                                                  const float* __restrict__ bsum,
                                                  const float* __restrict__ Wf,
                                                  const float* __restrict__ bf,
                                                  float* __restrict__ out, int nE) {
    const int e = blockIdx.x * (blockDim.x >> 5) + (threadIdx.x >> 5);
    if (e >= nE) return;
    const int lane = threadIdx.x & 31;
    const int a = ia[e], b = ib[e];
    const float4 p  = ((const float4*)(P + (size_t)a * D))[lane];
    const float4 q  = ((const float4*)(Q + (size_t)b * D))[lane];
    const float4 bb = ((const float4*)bsum)[lane];
    const float4 wf = ((const float4*)Wf)[lane];
    float acc = fmaxf(p.x + q.x + bb.x, 0.0f) * wf.x
              + fmaxf(p.y + q.y + bb.y, 0.0f) * wf.y
              + fmaxf(p.z + q.z + bb.z, 0.0f) * wf.z
              + fmaxf(p.w + q.w + bb.w, 0.0f) * wf.w;
    #pragma unroll
    for (int off = 16; off > 0; off >>= 1)
        acc += __shfl_xor(acc, off, 32);
    if (lane == 0)
        out[e] = 1.0f / (1.0f + __expf(-(acc + bf[0])));
}

// ---------------- host-side orchestration ----------------
extern "C" void kernel_launch(void* const* d_in, const int* in_sizes, int n_in,
                              void* d_out, int out_size, void* d_ws, size_t ws_size,
                              hipStream_t stream) {
    const float* node_feat = (const float*)d_in[0];
    const int*   src       = (const int*)d_in[1];
    const int*   dst       = (const int*)d_in[2];
    const int*   neg       = (const int*)d_in[3];
    const float* W1    = (const float*)d_in[4];
    const float* b1    = (const float*)d_in[5];
    const float* gamma = (const float*)d_in[6];
    const float* beta  = (const float*)d_in[7];
    const float* W2    = (const float*)d_in[8];
    const float* b2    = (const float*)d_in[9];
    const float* Ws    = (const float*)d_in[10];
    const float* bs    = (const float*)d_in[11];
    const float* Wd    = (const float*)d_in[12];
    const float* bd    = (const float*)d_in[13];
    const float* Wf    = (const float*)d_in[14];
    const float* bf    = (const float*)d_in[15];

    const int nNodes = in_sizes[0] / D;     // 100000 (divisible by 16)
    const int nE     = in_sizes[1];         // 1600000
    const long ND    = (long)nNodes * D;

    float* out = (float*)d_out;
    float* ws  = (float*)d_ws;

    // workspace layout (floats)
    float* dinv    = ws;                       // nNodes (used as deg first)
    float* stats   = dinv + nNodes;            // 2*D
    float* bnscale = stats + 2 * D;            // D
    float* bnshift = bnscale + D;              // D
    float* bsum    = bnshift + D;              // D
    float* bufA    = bsum + D;                 // ND
    float* bufB    = bufA + ND;                // ND
    float* bufC    = bufB + ND;                // ND

    const int zblocks    = 4096;
    const int ngemm      = nNodes / 16;                 // 6250 row tiles
    const int edgeBlocks = (nE + 7) / 8;                // 1 wave / edge, 8 waves / block
    const int nodeBlocks = (nNodes + 255) / 256;
    const int bnBlocks   = (nNodes + BN_ROWS - 1) / BN_ROWS;

    // degrees -> dinv  (also zero the stats that sit right after)
    gcn_zero_f32<<<64, 256, 0, stream>>>(dinv, (long)nNodes + 2 * D);
    gcn_degree<<<(nE + 255) / 256, 256, 0, stream>>>(dst, dinv, nE);
    gcn_make_dinv<<<nodeBlocks, 256, 0, stream>>>(dinv, nNodes);

    // layer 1: H1 = X @ W1 -> bufA ; aggregate -> bufB
    gcn_gemm128<<<ngemm, 256, 0, stream>>>(node_feat, W1, bufA);
    gcn_zero_f32<<<zblocks, 256, 0, stream>>>(bufB, ND);
    gcn_scatter<<<edgeBlocks, 256, 0, stream>>>(bufA, src, dst, dinv, bufB, nE);
    gcn_combine<<<zblocks, 256, 0, stream>>>(bufB, bufA, dinv, b1, bufB, ND);

    // batch norm + relu (in place on bufB)
    gcn_bn_stats<<<bnBlocks, 128, 0, stream>>>(bufB, stats, nNodes);
    gcn_bn_final<<<1, 128, 0, stream>>>(stats, gamma, beta, bs, bd,
                                        bnscale, bnshift, bsum, nNodes);
    gcn_bn_apply<<<zblocks, 256, 0, stream>>>(bufB, bnscale, bnshift, ND);

    // layer 2: H2 = x1 @ W2 -> bufA ; aggregate -> bufC = z
    gcn_gemm128<<<ngemm, 256, 0, stream>>>(bufB, W2, bufA);
    gcn_zero_f32<<<zblocks, 256, 0, stream>>>(bufC, ND);
    gcn_scatter<<<edgeBlocks, 256, 0, stream>>>(bufA, src, dst, dinv, bufC, nE);
    gcn_combine<<<zblocks, 256, 0, stream>>>(bufC, bufA, dinv, b2, bufC, ND);

    // decoder projections: P = z @ Ws -> bufA ; Q = z @ Wd -> bufB
    gcn_gemm128<<<ngemm, 256, 0, stream>>>(bufC, Ws, bufA);
    gcn_gemm128<<<ngemm, 256, 0, stream>>>(bufC, Wd, bufB);

    // per-edge decode: pos then neg
    gcn_decode<<<edgeBlocks, 256, 0, stream>>>(bufA, bufB, src, dst, bsum, Wf, bf, out, nE);
    gcn_decode<<<edgeBlocks, 256, 0, stream>>>(bufA, bufB, src, neg, bsum, Wf, bf, out + nE, nE);
}